// TTLinear_34961033789723
// MI455X (gfx1250) — compile-verified
//
#include <hip/hip_runtime.h>

// TT-linear on MI455X (gfx1250), wave32, fp32 WMMA (V_WMMA_F32_16X16X4_F32).
//
// x: (128, 8192) fp32, reshaped [t][i0=32][i1=16][i2=16]
// core0: (1,32,32,4)  c0[o0][i0][r1]
// core1: (4,16,16,4)  c1[r1][o1][i1][r2]
// core2: (4,16,16,1)  c2[r2][o2][i2]
// y[t, o0*256+o1*16+o2] = sum_{i0,i1,i2,r1,r2} x*c0*c1*c2 + bias
//
// Stage A+B fused kernel (grid = 8 token-tiles x 32 i0):
//   A: A1[t,i1,r2,o2]   = sum_{i2} x[t,i0,i1,i2] * c2[r2,o2,i2]        (LDS)
//   B: A2[t,o1,o2,i0,r1]= sum_{i1,r2} A1 * c1[r1,o1,i1,r2]             (-> ws)
// Stage C kernel:
//   y[t,o0,o1,o2] = sum_{i0,r1} A2[t,o1,o2,i0,r1] * c0[o0,i0,r1] + bias

typedef __attribute__((ext_vector_type(2))) float v2f;
typedef __attribute__((ext_vector_type(8))) float v8f;

__device__ __forceinline__ v8f wmma_f32(v2f a, v2f b, v8f c) {
  // D = A(16x4) * B(4x16) + C(16x16), full fp32
  return __builtin_amdgcn_wmma_f32_16x16x4_f32(
      /*neg_a=*/false, a, /*neg_b=*/false, b,
      /*c_mod=*/(short)0, c, /*reuse_a=*/false, /*reuse_b=*/false);
}

// A-frag (16x4 f32): lanes 0-15 -> M=lane, v0=K(kb+0), v1=K(kb+1);
//                    lanes 16-31 -> M=lane-16, K offset +2.
// B-frag (4x16 f32): lanes 0-15 -> N=lane, v0=K(kb+0) row, v1=K(kb+1);
//                    lanes 16-31 -> K offset +2.
// C/D (16x16 f32):   VGPR j: lanes 0-15 M=j, lanes 16-31 M=j+8; N=lane&15.

extern "C" __global__ void __launch_bounds__(256)
tt_stageAB(const float* __restrict__ x,
           const float* __restrict__ c1,
           const float* __restrict__ c2,
           float* __restrict__ A2)
{
  __shared__ float Xs[16 * 16 * 16];      // [t][i1][i2]              16 KB
  __shared__ float A1s[16 * 16 * 64];     // [(t*16+o2)][i1*4+r2]     64 KB
  __shared__ float C1s[64 * 64];          // [(r1*16+o1)][i1*4+r2]    16 KB
  __shared__ float C2s[4 * 16 * 16];      // [r2][o2][i2]              4 KB

  const int tid  = threadIdx.x;
  const int bt   = blockIdx.x >> 5;       // token tile 0..7
  const int bi0  = blockIdx.x & 31;       // i0
  const int lane = tid & 31;
  const int wave = tid >> 5;              // 0..7
  const int ml   = lane & 15;
  const int koff = (lane >> 4) << 1;      // 0 or 2 (K-half select)
  const int half = (lane >> 4) << 3;      // 0 or 8 (D M-half)

  // Stage weights into LDS (c1 flat == [(r1,o1)][(i1,r2)] row-major already)
  #pragma unroll
  for (int i = 0; i < 4; ++i)  C2s[tid + 256 * i] = c2[tid + 256 * i];
  #pragma unroll
  for (int i = 0; i < 16; ++i) C1s[tid + 256 * i] = c1[tid + 256 * i];
  // x slice for (bt, bi0): 16 tokens x 256 contiguous floats each
  #pragma unroll
  for (int i = 0; i < 16; ++i) {
    int idx = tid + 256 * i;              // t*256 + (i1*16+i2)
    int t = idx >> 8, f = idx & 255;
    Xs[idx] = x[(size_t)(bt * 16 + t) * 8192 + (size_t)bi0 * 256 + f];
  }
  __syncthreads();

  // ---- Stage A: M=(t,i1)=256, K=i2=16, N=o2=16, batched over r2 (4) ----
  #pragma unroll
  for (int j = 0; j < 8; ++j) {
    int job = wave * 8 + j;               // 64 jobs, 8 waves
    int mt = job >> 2;                    // M-tile 0..15
    int r2 = job & 3;
    v8f acc = {0.f, 0.f, 0.f, 0.f, 0.f, 0.f, 0.f, 0.f};
    #pragma unroll
    for (int ko = 0; ko < 4; ++ko) {      // K = 16 -> 4 wmma
      int kb = ko * 4 + koff;
      v2f a, b;
      a.x = Xs[(mt * 16 + ml) * 16 + kb];
      a.y = Xs[(mt * 16 + ml) * 16 + kb + 1];
      b.x = C2s[(r2 * 16 + ml) * 16 + kb];      // B[k][n] = c2[r2][n][k]
      b.y = C2s[(r2 * 16 + ml) * 16 + kb + 1];
      acc = wmma_f32(a, b, acc);
    }
    #pragma unroll
    for (int cmp = 0; cmp < 8; ++cmp) {   // D scatter -> A1s[(t,o2)][(i1,r2)]
      int m  = mt * 16 + cmp + half;      // (t,i1)
      int t  = m >> 4, i1 = m & 15;
      A1s[(t * 16 + ml) * 64 + i1 * 4 + r2] = acc[cmp];
    }
  }
  __syncthreads();

  // ---- Stage B: M=(t,o2)=256, K=(i1,r2)=64, N=(r1,o1)=64 ----
  #pragma unroll
  for (int j = 0; j < 8; ++j) {
    int job = wave * 8 + j;
    int mt = job >> 2;                    // M-tile 0..15
    int nt = job & 3;                     // N-tile 0..3
    v8f acc = {0.f, 0.f, 0.f, 0.f, 0.f, 0.f, 0.f, 0.f};
    #pragma unroll
    for (int ko = 0; ko < 16; ++ko) {     // K = 64 -> 16 wmma
      int kb = ko * 4 + koff;
      v2f a, b;
      a.x = A1s[(mt * 16 + ml) * 64 + kb];
      a.y = A1s[(mt * 16 + ml) * 64 + kb + 1];
      b.x = C1s[(nt * 16 + ml) * 64 + kb];      // B[kk][nn] = C1s[nn*64+kk]
      b.y = C1s[(nt * 16 + ml) * 64 + kb + 1];
      acc = wmma_f32(a, b, acc);
    }
    #pragma unroll
    for (int cmp = 0; cmp < 8; ++cmp) {
      int m2 = mt * 16 + cmp + half;      // t*16 + o2
      int t  = m2 >> 4, o2 = m2 & 15;
      int nn = nt * 16 + ml;              // r1*16 + o1
      int r1 = nn >> 4, o1 = nn & 15;
      size_t Tg = (size_t)(bt * 16 + t);
      // A2 layout [t][o1][o2][i0][r1] -> contiguous K rows for stage C
      A2[((Tg * 16 + o1) * 16 + o2) * 128 + bi0 * 4 + r1] = acc[cmp];
    }
  }
}

extern "C" __global__ void __launch_bounds__(256)
tt_stageC(const float* __restrict__ A2,
          const float* __restrict__ c0,
          const float* __restrict__ bias,
          float* __restrict__ out)
{
  __shared__ float C0s[32 * 128];         // [o0][i0*4+r1]  16 KB

  const int tid  = threadIdx.x;
  const int lane = tid & 31;
  const int wave = tid >> 5;
  const int ml   = lane & 15;
  const int koff = (lane >> 4) << 1;
  const int half = (lane >> 4) << 3;

  #pragma unroll
  for (int i = 0; i < 16; ++i) C0s[tid + 256 * i] = c0[tid + 256 * i];
  __syncthreads();

  // M=(t,o1,o2)=32768 rows, K=(i0,r1)=128, N=o0=32; 1 tile-pair per wave
  int job = blockIdx.x * 8 + wave;        // 0..4095
  int mt = job >> 1;                      // 0..2047
  int nt = job & 1;                       // 0..1

  v8f acc = {0.f, 0.f, 0.f, 0.f, 0.f, 0.f, 0.f, 0.f};
  const float* arow = A2 + (size_t)(mt * 16 + ml) * 128;
  #pragma unroll
  for (int ko = 0; ko < 32; ++ko) {       // K = 128 -> 32 wmma
    int kb = ko * 4 + koff;
    v2f a, b;
    a.x = arow[kb];
    a.y = arow[kb + 1];
    b.x = C0s[(nt * 16 + ml) * 128 + kb]; // B[kk][o0] = c0 flat [o0*128+kk]
    b.y = C0s[(nt * 16 + ml) * 128 + kb + 1];
    acc = wmma_f32(a, b, acc);
  }

  #pragma unroll
  for (int cmp = 0; cmp < 8; ++cmp) {
    int m = mt * 16 + cmp + half;         // t*256 + o1*16 + o2
    int n = nt * 16 + ml;                 // o0
    int t = m >> 8, rem = m & 255;
    out[(size_t)t * 8192 + n * 256 + rem] = acc[cmp] + bias[n * 256 + rem];
  }
}

extern "C" void kernel_launch(void* const* d_in, const int* in_sizes, int n_in,
                              void* d_out, int out_size, void* d_ws, size_t ws_size,
                              hipStream_t stream) {
  const float* x    = (const float*)d_in[0];
  const float* c0   = (const float*)d_in[1];
  const float* c1   = (const float*)d_in[2];
  const float* c2   = (const float*)d_in[3];
  const float* bias = (const float*)d_in[4];
  float* out = (float*)d_out;
  float* A2  = (float*)d_ws;              // 128*16*16*128 floats = 16 MB

  tt_stageAB<<<dim3(8 * 32), dim3(256), 0, stream>>>(x, c1, c2, A2);
  tt_stageC <<<dim3(512),    dim3(256), 0, stream>>>(A2, c0, bias, out);
}